// HungarianMatcher_4818953306569
// MI455X (gfx1250) — compile-verified
//
#include <hip/hip_runtime.h>
#include <math.h>

// ---------------- constants matching the reference ----------------
#define K_CLASS 1.0f
#define K_BBOX  5.0f
#define K_GIOU  2.0f
#define EPS_F   1e-6f
#define INF_F   1e30f

#define TGT_TILE 16           // targets staged per block via TDM
#define MCAP 2048             // max columns (preds) for JV LDS state
#define NCAP 512              // max rows (targets)

typedef unsigned int u32x4 __attribute__((ext_vector_type(4)));
typedef int          i32x4 __attribute__((ext_vector_type(4)));
typedef int          i32x8 __attribute__((ext_vector_type(8)));

// =================================================================
// Kernel 1: cost matrix, stored transposed CT[M][N] (targets-major)
// Target-box tile staged into LDS with the Tensor Data Mover.
// =================================================================
__global__ __launch_bounds__(256) void hm_cost_kernel(
    const float4* __restrict__ pred,   // N x (cx,cy,w,h)
    const float*  __restrict__ logit,  // N
    const float*  __restrict__ tgtF,   // M x 4 flat
    float* __restrict__ CT,            // M x N
    int N, int M)
{
  __shared__ float sTgt[TGT_TILE * 4];
  const int tid   = threadIdx.x;
  const int t0    = blockIdx.y * TGT_TILE;
  const int nElem = TGT_TILE * 4;      // 64 f32 = 256 B tile

#if __has_builtin(__builtin_amdgcn_tensor_load_to_lds) && __has_builtin(__builtin_amdgcn_s_wait_tensorcnt)
  // TDM issues regardless of EXEC, so gate on a *wave-uniform* branch:
  // only wave 0 of the block builds the descriptor and fires the DMA.
  if ((tid >> 5) == 0) {
    unsigned lds_off = (unsigned)(size_t)(void*)&sTgt[0];       // low 32b = LDS byte offset
    unsigned long long ga = (unsigned long long)(const void*)(tgtF + (size_t)t0 * 4);
    // D# group0: count=1, lds_addr, global_addr[56:0], type=2 ("image")
    u32x4 g0;
    g0[0] = 1u;
    g0[1] = lds_off;
    g0[2] = (unsigned)(ga & 0xFFFFFFFFull);
    g0[3] = (unsigned)((ga >> 32) & 0x1FFFFFFull) | (2u << 30);
    // D# group1: data_size=4B, 1-D tile: tensor_dim0 = tile_dim0 = 64 elems,
    // tensor_dim1 = tile_dim1 = 1, tensor_dim0_stride = 64.
    i32x8 g1;
    g1[0] = 0x00020000;                       // workgroup_mask=0, data_size=2 (4B)
    g1[1] = (nElem & 0xFFFF) << 16;           // tensor_dim0[15:0] @ bits 63:48
    g1[2] = 1 << 16;                          // tensor_dim0 hi=0 | tensor_dim1 lo=1
    g1[3] = (nElem & 0xFFFF) << 16;           // tensor_dim1 hi=0 | tile_dim0=64
    g1[4] = 1;                                // tile_dim1=1, tile_dim2=0
    g1[5] = nElem;                            // tensor_dim0_stride[31:0]
    g1[6] = 0;
    g1[7] = 0;
    i32x4 gz; gz[0] = 0; gz[1] = 0; gz[2] = 0; gz[3] = 0;   // groups 2/3 unused (2-D)
#if defined(__clang_major__) && (__clang_major__ >= 23)
    i32x8 gz8; for (int q = 0; q < 8; ++q) gz8[q] = 0;
    __builtin_amdgcn_tensor_load_to_lds(g0, g1, gz, gz, gz8, 0);
#else
    __builtin_amdgcn_tensor_load_to_lds(g0, g1, gz, gz, 0);
#endif
    __builtin_amdgcn_s_wait_tensorcnt(0);
  }
#else
  if (tid < nElem) sTgt[tid] = tgtF[(size_t)t0 * 4 + tid];
#endif
  __syncthreads();

  const int i = blockIdx.x * blockDim.x + tid;
  if (i >= N) return;

  const float4 pb = pred[i];
  const float prob = 1.0f / (1.0f + __expf(-logit[i]));
  const float px1 = pb.x - 0.5f * pb.z, py1 = pb.y - 0.5f * pb.w;
  const float px2 = pb.x + 0.5f * pb.z, py2 = pb.y + 0.5f * pb.w;
  const float parea = (px2 - px1) * (py2 - py1);

#pragma unroll
  for (int t = 0; t < TGT_TILE; ++t) {
    const float tcx = sTgt[t * 4 + 0], tcy = sTgt[t * 4 + 1];
    const float tw  = sTgt[t * 4 + 2], th  = sTgt[t * 4 + 3];
    const float l1 = fabsf(pb.x - tcx) + fabsf(pb.y - tcy) +
                     fabsf(pb.z - tw ) + fabsf(pb.w - th );
    const float tx1 = tcx - 0.5f * tw, ty1 = tcy - 0.5f * th;
    const float tx2 = tcx + 0.5f * tw, ty2 = tcy + 0.5f * th;
    const float ix1 = fmaxf(px1, tx1), iy1 = fmaxf(py1, ty1);
    const float ix2 = fminf(px2, tx2), iy2 = fminf(py2, ty2);
    const float inter = fmaxf(ix2 - ix1, 0.0f) * fmaxf(iy2 - iy1, 0.0f);
    const float tarea = (tx2 - tx1) * (ty2 - ty1);
    const float uni   = parea + tarea - inter;
    const float iou   = inter / (uni + EPS_F);
    const float ex1 = fminf(px1, tx1), ey1 = fminf(py1, ty1);
    const float ex2 = fmaxf(px2, tx2), ey2 = fmaxf(py2, ty2);
    const float enc = (ex2 - ex1) * (ey2 - ey1);
    const float giou = iou - (enc - uni) / (enc + EPS_F);
    CT[(size_t)(t0 + t) * N + i] = -K_CLASS * prob + K_BBOX * l1 - K_GIOU * giou;
  }
}

// =================================================================
// Kernel 2: Jonker-Volgenant LSA. One persistent 1024-thread block
// (32 wave32 waves). All solver state LDS-resident (~50 KB of the
// 320 KB/WGP). Argmin via wave32 shfl_xor lexicographic reduce.
// n = rows (targets, 512), m = cols (preds, 2048); CT is n x m.
// =================================================================
__global__ __launch_bounds__(1024) void hm_jv_kernel(
    const float* __restrict__ CT, float* __restrict__ out, int n, int m)
{
  __shared__ float sMinv[MCAP + 1];
  __shared__ float sV[MCAP + 1];
  __shared__ int   sWay[MCAP + 1];
  __shared__ int   sUsed[MCAP + 1];
  __shared__ int   sP[MCAP + 1];
  __shared__ float sU[NCAP + 1];
  __shared__ float sRedV[32];
  __shared__ int   sRedJ[32];
  __shared__ int   sGrp[MCAP / 32];
  __shared__ float sDelta;
  __shared__ int   sJ1, sJ0, sI0;

  const int tid  = threadIdx.x;
  const int lane = tid & 31;
  const int wave = tid >> 5;

  for (int j = tid; j <= m; j += 1024) { sV[j] = 0.0f; sP[j] = 0; sWay[j] = 0; }
  for (int r = tid; r <= n; r += 1024) sU[r] = 0.0f;
  __syncthreads();

  for (int i = 1; i <= n; ++i) {
    for (int j = tid; j <= m; j += 1024) { sMinv[j] = INF_F; sUsed[j] = 0; }
    if (tid == 0) { sP[0] = i; sJ0 = 0; }
    __syncthreads();

    while (true) {
      if (tid == 0) { sUsed[sJ0] = 1; sI0 = sP[sJ0]; }
      __syncthreads();
      const int   j0c = sJ0;
      const int   i0  = sI0;
      const float ui0 = sU[i0];
      const float* __restrict__ row = CT + (size_t)(i0 - 1) * m;

      // scan free columns: relax minv/way, collect local argmin
      float bv = INF_F; int bj = 0;
      for (int j = tid + 1; j <= m; j += 1024) {
        if (!sUsed[j]) {
          const float cur = row[j - 1] - ui0 - sV[j];
          float mv = sMinv[j];
          if (cur < mv) { mv = cur; sMinv[j] = cur; sWay[j] = j0c; }
          if (mv < bv || (mv == bv && j < bj)) { bv = mv; bj = j; }
        }
      }
      // wave32 lexicographic (val, idx) argmin
      for (int off = 16; off > 0; off >>= 1) {
        const float ov = __shfl_xor(bv, off, 32);
        const int   oj = __shfl_xor(bj, off, 32);
        if (ov < bv || (ov == bv && oj < bj)) { bv = ov; bj = oj; }
      }
      if (lane == 0) { sRedV[wave] = bv; sRedJ[wave] = bj; }
      __syncthreads();
      if (wave == 0) {
        bv = sRedV[lane]; bj = sRedJ[lane];
        for (int off = 16; off > 0; off >>= 1) {
          const float ov = __shfl_xor(bv, off, 32);
          const int   oj = __shfl_xor(bj, off, 32);
          if (ov < bv || (ov == bv && oj < bj)) { bv = ov; bj = oj; }
        }
        if (lane == 0) { sDelta = bv; sJ1 = bj; }
      }
      __syncthreads();
      const float delta = sDelta;
      const int   j1    = sJ1;

      // dual updates (used columns carry distinct rows -> race-free)
      for (int j = tid; j <= m; j += 1024) {
        if (sUsed[j]) { sU[sP[j]] += delta; sV[j] -= delta; }
        else          { sMinv[j] -= delta; }
      }
      if (tid == 0) sJ0 = j1;
      __syncthreads();
      if (sP[j1] == 0) break;   // uniform: all threads read same LDS value
    }

    if (tid == 0) {              // augment along 'way' (short path)
      int j0 = sJ0;
      while (j0) { const int jn = sWay[j0]; sP[j0] = sP[jn]; j0 = jn; }
    }
    __syncthreads();
  }

  // ---- compaction: emit (pred_idx, target_idx) ascending by pred ----
  const int nGroups = m >> 5;
  const int passes  = (m + 1023) >> 10;
  for (int p = 0; p < passes; ++p) {
    const int j = p * 1024 + tid + 1;
    const int flag = (j <= m) && (sP[j] != 0);
    const unsigned mask = (unsigned)__ballot(flag);
    const int g = p * 32 + wave;
    if (lane == 0 && g < nGroups) sGrp[g] = __popc(mask);
  }
  __syncthreads();
  if (tid == 0) {
    int acc = 0;
    for (int g = 0; g < nGroups; ++g) { const int c = sGrp[g]; sGrp[g] = acc; acc += c; }
  }
  __syncthreads();
  for (int p = 0; p < passes; ++p) {
    const int j = p * 1024 + tid + 1;
    const int flag = (j <= m) && (sP[j] != 0);
    const unsigned mask = (unsigned)__ballot(flag);
    const int g = p * 32 + wave;
    if (flag && g < nGroups) {
      const int pos = sGrp[g] + __popc(mask & ((1u << lane) - 1u));
      out[pos]     = (float)(j - 1);       // pred index
      out[n + pos] = (float)(sP[j] - 1);   // target index
    }
  }
}

// =================================================================
extern "C" void kernel_launch(void* const* d_in, const int* in_sizes, int n_in,
                              void* d_out, int out_size, void* d_ws, size_t ws_size,
                              hipStream_t stream) {
  (void)n_in; (void)out_size; (void)ws_size;
  const float* pred_boxes   = (const float*)d_in[0];   // (N,4)
  const float* pred_logits  = (const float*)d_in[1];   // (N,)
  const float* target_boxes = (const float*)d_in[2];   // (M,4)
  const int N = in_sizes[0] / 4;
  const int M = in_sizes[2] / 4;

  float* CT = (float*)d_ws;   // M x N transposed cost matrix (4 MB)

  dim3 grid((N + 255) / 256, (M + TGT_TILE - 1) / TGT_TILE);
  hm_cost_kernel<<<grid, 256, 0, stream>>>(
      (const float4*)pred_boxes, pred_logits, target_boxes, CT, N, M);

  hm_jv_kernel<<<1, 1024, 0, stream>>>(CT, (float*)d_out, M, N);
}